// S4DKernel_2817498546781
// MI455X (gfx1250) — compile-verified
//
#include <hip/hip_runtime.h>
#include <hip/hip_bf16.h>

typedef __attribute__((ext_vector_type(2))) float v2f;
typedef __attribute__((ext_vector_type(8))) float v8f;

#define NSTATE 64

// out[l] = sum_n wr[n] * exp(dtA[n]*l)   recast as a (1024x64)x(64x16) f32 GEMM
// via l = 16*lo + li, executed on v_wmma_f32_16x16x4_f32.
// One wave (32 threads) per 16x16 output tile = 256 consecutive l values.
__global__ __launch_bounds__(32) void s4d_vandermonde_wmma(
    const float* __restrict__ log_dt,
    const float* __restrict__ log_A_real,
    const float* __restrict__ A_imag,
    const float* __restrict__ Bv,
    const float* __restrict__ Cv,
    float* __restrict__ out,
    int L)
{
    __shared__ float s_wr[NSTATE];
    __shared__ float s_dtA[NSTATE];

    const int lane = threadIdx.x & 31;
    const int tile = blockIdx.x;              // tile covers l in [256*tile, 256*tile+256)

    const float dt = __expf(log_dt[0]);

    // Precompute per-state weight and decay into LDS (2 states per lane).
    for (int n = lane; n < NSTATE; n += 32) {
        float ar   = -__expf(log_A_real[n]);          // Re(A)
        float ai   = A_imag[n];                       // Im(A)
        float dtA  = ar * dt;                         // real (reference uses Re(A) only)
        float edtA = __expf(dtA);
        // wr = Re( B*C*(edtA-1)/A ) = B*C*(edtA-1)*ar / (ar^2 + ai^2)
        float wr = Bv[n] * Cv[n] * (edtA - 1.0f) * ar / (ar * ar + ai * ai);
        s_dtA[n] = dtA;
        s_wr[n]  = wr;
    }
    __syncthreads();

    // Fragment coordinates (ISA 7.12.2 layouts for 32-bit WMMA operands):
    //   A 16x4 : lane holds row M = lane&15, K = 2*(lane>>4) + j   (j = vgpr 0/1)
    //   B 4x16 : lane holds col N = lane&15, K = 2*(lane>>4) + j
    //   C/D    : lane 0-15 -> N=lane, VGPR v -> M=v ; lanes 16-31 -> M=v+8
    const int   g   = lane >> 4;                  // half-wave selector
    const int   rc  = lane & 15;                  // M (A rows) == N (B cols)
    const float loG = 16.0f * (float)(16 * tile + rc);  // A row exponent multiplier
    const float li  = (float)rc;                        // B col exponent multiplier

    v8f acc = {};
    #pragma unroll
    for (int kb = 0; kb < 16; ++kb) {             // K = 64 in chunks of 4
        const int n0 = 4 * kb + 2 * g;
        v2f a, b;
        a.x = s_wr[n0]     * __expf(s_dtA[n0]     * loG);
        a.y = s_wr[n0 + 1] * __expf(s_dtA[n0 + 1] * loG);
        b.x = __expf(s_dtA[n0]     * li);
        b.y = __expf(s_dtA[n0 + 1] * li);
        // 8 args: (neg_a, A, neg_b, B, c_mod, C, reuse_a, reuse_b)
        acc = __builtin_amdgcn_wmma_f32_16x16x4_f32(false, a, false, b,
                                                    (short)0, acc, false, false);
    }

    // Scatter D: l = 256*tile + 16*M + N, M = v + 8*g, N = rc.
    float* outp = out + tile * 256;
    #pragma unroll
    for (int v = 0; v < 8; ++v) {
        int l = 16 * (v + 8 * g) + rc;
        if (tile * 256 + l < L) outp[l] = acc[v];
    }
}

extern "C" void kernel_launch(void* const* d_in, const int* in_sizes, int n_in,
                              void* d_out, int out_size, void* d_ws, size_t ws_size,
                              hipStream_t stream) {
    (void)in_sizes; (void)n_in; (void)d_ws; (void)ws_size;
    const float* log_dt     = (const float*)d_in[0];
    const float* log_A_real = (const float*)d_in[1];
    const float* A_imag     = (const float*)d_in[2];
    const float* B          = (const float*)d_in[3];
    const float* C          = (const float*)d_in[4];
    // d_in[5]/d_in[6] (state_real/state_imag) are dead: reference returns K[-1],
    // the Bc row — all state-derived rows are discarded. d_in[7] is L; its value
    // equals out_size since H == 1.
    float* out = (float*)d_out;

    const int L = out_size;                 // H == 1 -> out is (1, L)
    const int tiles = (L + 255) / 256;      // 64 tiles for L = 16384
    s4d_vandermonde_wmma<<<tiles, 32, 0, stream>>>(log_dt, log_A_real, A_imag,
                                                   B, C, out, L);
}